// LR2_5403068859012
// MI455X (gfx1250) — compile-verified
//
#include <hip/hip_runtime.h>

typedef float v2f __attribute__((ext_vector_type(2)));
typedef float v8f __attribute__((ext_vector_type(8)));

#define LENGTH 200
#define BATCH  1024
#define TILE_N 16
#define THREADS 256

__global__ __launch_bounds__(THREADS)
void presence_linear_wmma_kernel(const unsigned long long* __restrict__ x,
                                 const float* __restrict__ W,
                                 const float* __restrict__ b_lut,
                                 const float* __restrict__ bias,
                                 float* __restrict__ out) {
    // 200x16 token tile + 200x16 masked-value tile: 25.6 KB LDS
    __shared__ unsigned int toks[LENGTH * TILE_N];
    __shared__ float        vals[LENGTH * TILE_N];

    const int tid = threadIdx.x;
    const int n0  = blockIdx.x * TILE_N;

    // ---- Phase 1: stage token tile into LDS (coalesced 128B rows) ----
    for (int idx = tid; idx < LENGTH * TILE_N; idx += THREADS) {
        const int i = idx >> 4;   // row (position in sequence)
        const int n = idx & 15;   // column within tile
        toks[idx] = (unsigned int)x[(size_t)i * BATCH + (n0 + n)];
    }
    __syncthreads();

    // ---- Phase 2: first-occurrence mask (dedup) + W gather ----
    // masked[i][n] = (token first seen at row i in column n) ? W[token] : 0
    for (int idx = tid; idx < LENGTH * TILE_N; idx += THREADS) {
        const int i = idx >> 4;
        const int n = idx & 15;
        const unsigned int tok = toks[idx];
        int dup = 0;
        for (int j = 0; j < i; ++j) {
            dup |= (toks[(j << 4) + n] == tok);
        }
        vals[idx] = dup ? 0.0f : W[tok];
    }
    __syncthreads();

    // ---- Phase 3: wave 0 reduces the 200x16 tile with V_WMMA_F32_16X16X4_F32
    // A = ones(16x4)  =>  D[m][n] = sum_k B[k][n] + C[m][n]  (every row = colsum)
    // B layout (32-bit, 4x16 K x N): lane = n + 16*(k>>1), vgpr = k&1
    if (tid < 32) {
        const int lane = tid;
        const int n    = lane & 15;
        const int kh   = lane >> 4;          // which K half-pair this lane holds

        v2f a;
        a.x = 1.0f;
        a.y = 1.0f;
        v8f c = {};

        for (int kb = 0; kb < LENGTH; kb += 4) {   // 50 WMMA accumulations
            v2f b;
            b.x = vals[(kb + 2 * kh + 0) * TILE_N + n];
            b.y = vals[(kb + 2 * kh + 1) * TILE_N + n];
            c = __builtin_amdgcn_wmma_f32_16x16x4_f32(
                    /*neg_a=*/false, a,
                    /*neg_b=*/false, b,
                    /*c_mod=*/(short)0, c,
                    /*reuse_a=*/false, /*reuse_b=*/false);
        }

        // D row M=0 lives in VGPR0 of lanes 0..15 (lane = N)
        if (lane < 16) {
            out[n0 + lane] = c[0] + b_lut[0] + bias[0];
        }
    }
}

extern "C" void kernel_launch(void* const* d_in, const int* in_sizes, int n_in,
                              void* d_out, int out_size, void* d_ws, size_t ws_size,
                              hipStream_t stream) {
    (void)in_sizes; (void)n_in; (void)out_size; (void)d_ws; (void)ws_size;
    const unsigned long long* x     = (const unsigned long long*)d_in[0]; // int64 tokens
    const float*              W     = (const float*)d_in[1];              // (1, VSIZE)
    const float*              b_lut = (const float*)d_in[2];              // (1,)
    const float*              bias  = (const float*)d_in[3];              // (1,)
    float*                    out   = (float*)d_out;                      // (BATCH,)

    dim3 grid(BATCH / TILE_N);   // 64 workgroups, one 16-column tile each
    presence_linear_wmma_kernel<<<grid, THREADS, 0, stream>>>(x, W, b_lut, bias, out);
}